// RSHN_58342835749536
// MI455X (gfx1250) — compile-verified
//
#include <hip/hip_runtime.h>
#include <hip/hip_bf16.h>

typedef __attribute__((ext_vector_type(2))) float v2f;
typedef __attribute__((ext_vector_type(8))) float v8f;

#define D 128
#define LDST 130   // padded LDS row stride (floats); 520B keeps 8B alignment for ds_load_b64

// Guaranteed-hardware f32 atomic add (global_atomic_add_f32, no-return when
// the result is unused).  A generic atomicrmw fadd can be expanded to a CAS
// loop by the backend for fine-grained-memory safety; unsafeAtomicAdd opts out.
__device__ __forceinline__ void atomic_add_f32(float* p, float v) {
#if defined(__HIP_DEVICE_COMPILE__) && defined(__AMDGCN__)
    (void)unsafeAtomicAdd(p, v);
#else
    (void)atomicAdd(p, v);
#endif
}

// ---------------------------------------------------------------------------
// Phase 1: tiny relation encoder over the coarsened line graph (R=3, E_CL=6).
// One block of 128 threads; thread t owns column t of all 3 rows.
// ---------------------------------------------------------------------------
__global__ void rshn_relation_encoder(const float* __restrict__ cl_h,
                                      const float* __restrict__ cl_w,
                                      const float* __restrict__ beta1,
                                      const float* __restrict__ beta2,
                                      const float* __restrict__ W_e1,
                                      const float* __restrict__ b_e1,
                                      const int* __restrict__ cl_src,
                                      const int* __restrict__ cl_dst,
                                      float* __restrict__ h_rel) {
    const int t = threadIdx.x;           // 0..127
    __shared__ float sh[3][D];
    __shared__ float ssq[3][D];

    float h[3];
#pragma unroll
    for (int r = 0; r < 3; ++r) h[r] = cl_h[r * D + t];

    float betas[2];
    betas[0] = beta1[0];
    betas[1] = beta2[0];

    int es[6], ed[6];
    float ew[6];
#pragma unroll
    for (int e = 0; e < 6; ++e) { es[e] = cl_src[e]; ed[e] = cl_dst[e]; ew[e] = cl_w[e]; }

    for (int layer = 0; layer < 2; ++layer) {
#pragma unroll
        for (int r = 0; r < 3; ++r) { sh[r][t] = h[r]; ssq[r][t] = h[r] * h[r]; }
        __syncthreads();

        // L2 norms of each row (redundant per-thread reduction: tiny)
        float nrm[3];
#pragma unroll
        for (int r = 0; r < 3; ++r) {
            float s = 0.f;
            for (int k = 0; k < D; ++k) s += ssq[r][k];
            nrm[r] = fmaxf(sqrtf(s), 1e-12f);
        }

        // edge softmax normalized by src (6 edges, computed redundantly per thread)
        const float beta = betas[layer];
        float emax[3] = {-1e30f, -1e30f, -1e30f};
#pragma unroll
        for (int e = 0; e < 6; ++e) {
            float v = beta * ew[e];
            if (v > emax[es[e]]) emax[es[e]] = v;
        }
        float p[6];
        float psum[3] = {0.f, 0.f, 0.f};
#pragma unroll
        for (int e = 0; e < 6; ++e) {
            p[e] = expf(beta * ew[e] - emax[es[e]]);
            psum[es[e]] += p[e];
        }
#pragma unroll
        for (int e = 0; e < 6; ++e) p[e] /= psum[es[e]];

        float agg[3] = {0.f, 0.f, 0.f};
#pragma unroll
        for (int e = 0; e < 6; ++e)
            agg[ed[e]] += (sh[es[e]][t] / nrm[es[e]]) * p[e];

#pragma unroll
        for (int r = 0; r < 3; ++r) h[r] = fmaxf(h[r] + agg[r], 0.f);  // relu(h + agg)
        __syncthreads();
    }

#pragma unroll
    for (int r = 0; r < 3; ++r) sh[r][t] = h[r];
    __syncthreads();

    // h_rel = h @ W_e1 + b_e1   (coalesced along t)
#pragma unroll
    for (int r = 0; r < 3; ++r) {
        float acc = b_e1[t];
        for (int k = 0; k < D; ++k) acc += sh[r][k] * W_e1[k * D + t];
        h_rel[r * D + t] = acc;
    }
}

// ---------------------------------------------------------------------------
// Zero-fill (agg buffers must be re-zeroed every call; harness does not).
// ---------------------------------------------------------------------------
__global__ void rshn_zero(float* __restrict__ p, long n) {
    long i = (long)blockIdx.x * blockDim.x + threadIdx.x;
    long stride = (long)gridDim.x * blockDim.x;
    for (; i < n; i += stride) p[i] = 0.f;
}

// ---------------------------------------------------------------------------
// Phase 2: edge scatter.  One wave32 per edge: lane l handles columns
// [4l, 4l+4): a coalesced 512B load of the src row, multiply by the
// per-relation weight vector, 4 no-return hardware f32 atomic adds.
// ---------------------------------------------------------------------------
__global__ __launch_bounds__(256) void rshn_scatter(const float* __restrict__ feat,
                                                    const int* __restrict__ src,
                                                    const int* __restrict__ dst,
                                                    const float* __restrict__ wrel,
                                                    float* __restrict__ agg,
                                                    int nEdges, int maskSelf) {
    const int lane = threadIdx.x & 31;
    const int wave = threadIdx.x >> 5;
    const int e = blockIdx.x * 8 + wave;
    if (e >= nEdges) return;                 // uniform per wave

    const int s = src[e];
    const int d = dst[e];
    if (maskSelf && (s == d)) return;        // drop AA self-loops

    const float4 f4 = *(const float4*)(feat + (size_t)s * D + lane * 4);
    const float4 w4 = *(const float4*)(wrel + lane * 4);
    float* ap = agg + (size_t)d * D + lane * 4;

    atomic_add_f32(ap + 0, f4.x * w4.x);
    atomic_add_f32(ap + 1, f4.y * w4.y);
    atomic_add_f32(ap + 2, f4.z * w4.z);
    atomic_add_f32(ap + 3, f4.w * w4.w);
}

// ---------------------------------------------------------------------------
// Phase 3: fused GEMM  out = act((agg + f) @ W)  with W 128x128 f32.
// Block = 256 threads = 8 waves, 128 rows per block.
// W is staged transposed in LDS; X = agg + f staged row-major in LDS.
// Each wave: 16-row strip, 8 N-tiles, 32 K-steps of v_wmma_f32_16x16x4_f32.
//
// f32 16x16x4 WMMA register layout (ISA 7.12.2):
//   A (16x4): lane l(0..15) holds A[l][0],A[l][1]; lane l+16 holds A[l][2],A[l][3]
//   B (4x16): mirrored with N across lanes, K across the v2f elements
//   C/D (16x16): lane l(0..15) N=l, vgpr v -> M=v; lanes 16..31 -> M=v+8
// ---------------------------------------------------------------------------
template <bool ADD_AGG, bool APPLY_TANH>
__global__ __launch_bounds__(256) void rshn_gemm(const float* __restrict__ agg,
                                                 const float* __restrict__ f,
                                                 const float* __restrict__ W,
                                                 float* __restrict__ out, int M) {
    __shared__ float Xlds[128 * LDST];
    __shared__ float WT[D * LDST];

    const int tid = threadIdx.x;
    const int base = blockIdx.x * 128;

    // Stage W transposed: WT[n][k] = W[k][n]; coalesced global reads,
    // consecutive-n LDS writes at stride LDST (conflict-free).
    for (int idx = tid; idx < D * D; idx += 256) {
        const int k = idx >> 7;
        const int n = idx & (D - 1);
        WT[n * LDST + k] = W[idx];
    }

    // Stage X = agg + f (or just f), float4-coalesced; zero-pad rows >= M.
    {
        const int cg = tid & 31;   // column group: 4 floats each
        const int r0 = tid >> 5;   // 0..7
        for (int rr = r0; rr < 128; rr += 8) {
            const int row = base + rr;
            float4 x = make_float4(0.f, 0.f, 0.f, 0.f);
            if (row < M) {
                const float4 fv = *(const float4*)(f + (size_t)row * D + cg * 4);
                if (ADD_AGG) {
                    const float4 av = *(const float4*)(agg + (size_t)row * D + cg * 4);
                    x = make_float4(fv.x + av.x, fv.y + av.y, fv.z + av.z, fv.w + av.w);
                } else {
                    x = fv;
                }
            }
            *(float4*)&Xlds[rr * LDST + cg * 4] = x;
        }
    }
    __syncthreads();

    const int lane = tid & 31;
    const int wave = tid >> 5;
    const int lmod = lane & 15;
    const int hi   = lane >> 4;   // 0: K=0,1  1: K=2,3 halves of the fragment

    v8f acc[8] = {};  // 8 N-tiles of 16x16 f32 accumulators

    const float* xrow = &Xlds[(wave * 16 + lmod) * LDST];
#pragma unroll 4
    for (int ks = 0; ks < 32; ++ks) {
        const v2f a = *(const v2f*)(xrow + ks * 4 + hi * 2);
#pragma unroll
        for (int nt = 0; nt < 8; ++nt) {
            const v2f b = *(const v2f*)(&WT[(nt * 16 + lmod) * LDST + ks * 4 + hi * 2]);
            acc[nt] = __builtin_amdgcn_wmma_f32_16x16x4_f32(
                false, a, false, b, (short)0, acc[nt], false, false);
        }
    }

    // Store: lane lmod holds column nt*16+lmod, vgpr v holds row v + 8*hi.
#pragma unroll
    for (int nt = 0; nt < 8; ++nt) {
#pragma unroll
        for (int v = 0; v < 8; ++v) {
            const int row = base + wave * 16 + v + 8 * hi;
            if (row < M) {
                float val = acc[nt][v];
                if (APPLY_TANH) val = tanhf(val);
                out[(size_t)row * D + nt * 16 + lmod] = val;
            }
        }
    }
}

// ---------------------------------------------------------------------------
// Host-side orchestration (graph-capture safe: only kernel launches on stream)
// ---------------------------------------------------------------------------
extern "C" void kernel_launch(void* const* d_in, const int* in_sizes, int n_in,
                              void* d_out, int out_size, void* d_ws, size_t ws_size,
                              hipStream_t stream) {
    const float* cl_h   = (const float*)d_in[0];
    const float* cl_w   = (const float*)d_in[1];
    const float* beta1  = (const float*)d_in[2];
    const float* beta2  = (const float*)d_in[3];
    const float* W_e1   = (const float*)d_in[4];
    const float* b_e1   = (const float*)d_in[5];
    const float* feat_A = (const float*)d_in[6];
    const float* feat_B = (const float*)d_in[7];
    const float* Wg1    = (const float*)d_in[8];
    const float* Wg2    = (const float*)d_in[9];
    const float* W_out  = (const float*)d_in[10];
    const int* cl_src   = (const int*)d_in[11];
    const int* cl_dst   = (const int*)d_in[12];
    const int* src_AA   = (const int*)d_in[13];
    const int* dst_AA   = (const int*)d_in[14];
    const int* src_AB   = (const int*)d_in[15];
    const int* dst_AB   = (const int*)d_in[16];
    const int* src_BA   = (const int*)d_in[17];
    const int* dst_BA   = (const int*)d_in[18];

    const int NA   = in_sizes[6] / D;
    const int NB   = in_sizes[7] / D;
    const int E_AA = in_sizes[13];
    const int E_AB = in_sizes[15];
    const int E_BA = in_sizes[17];

    // Workspace layout (floats): h_rel | aggA | aggB | fA1 | fB1
    float* ws    = (float*)d_ws;
    float* h_rel = ws;                                   // 3*128, padded to 1024
    float* aggA  = ws + 1024;
    float* aggB  = aggA + (size_t)NA * D;
    float* fA1   = aggB + (size_t)NB * D;
    float* fB1   = fA1 + (size_t)NA * D;

    const float* w_AA = h_rel + 0 * D;
    const float* w_AB = h_rel + 1 * D;
    const float* w_BA = h_rel + 2 * D;

    const int gemmA = (NA + 127) / 128;
    const int gemmB = (NB + 127) / 128;
    const int eblkAA = (E_AA + 7) / 8;
    const int eblkAB = (E_AB + 7) / 8;
    const int eblkBA = (E_BA + 7) / 8;

    // --- relation encoder ---
    rshn_relation_encoder<<<1, 128, 0, stream>>>(cl_h, cl_w, beta1, beta2, W_e1, b_e1,
                                                 cl_src, cl_dst, h_rel);

    // --- layer 1 ---
    rshn_zero<<<2048, 256, 0, stream>>>(aggA, (long)NA * D);
    rshn_zero<<<2048, 256, 0, stream>>>(aggB, (long)NB * D);
    rshn_scatter<<<eblkAA, 256, 0, stream>>>(feat_A, src_AA, dst_AA, w_AA, aggA, E_AA, 1);
    rshn_scatter<<<eblkBA, 256, 0, stream>>>(feat_B, src_BA, dst_BA, w_BA, aggA, E_BA, 0);
    rshn_scatter<<<eblkAB, 256, 0, stream>>>(feat_A, src_AB, dst_AB, w_AB, aggB, E_AB, 0);
    rshn_gemm<true, true><<<gemmA, 256, 0, stream>>>(aggA, feat_A, Wg1, fA1, NA);
    rshn_gemm<true, true><<<gemmB, 256, 0, stream>>>(aggB, feat_B, Wg1, fB1, NB);

    // --- layer 2 (GEMM output aliases agg buffers: row-disjoint blocks,
    //     each block stages its rows into LDS before any store) ---
    rshn_zero<<<2048, 256, 0, stream>>>(aggA, (long)NA * D);
    rshn_zero<<<2048, 256, 0, stream>>>(aggB, (long)NB * D);
    rshn_scatter<<<eblkAA, 256, 0, stream>>>(fA1, src_AA, dst_AA, w_AA, aggA, E_AA, 1);
    rshn_scatter<<<eblkBA, 256, 0, stream>>>(fB1, src_BA, dst_BA, w_BA, aggA, E_BA, 0);
    rshn_scatter<<<eblkAB, 256, 0, stream>>>(fA1, src_AB, dst_AB, w_AB, aggB, E_AB, 0);
    rshn_gemm<true, true><<<gemmA, 256, 0, stream>>>(aggA, fA1, Wg2, aggA, NA);
    rshn_gemm<true, true><<<gemmB, 256, 0, stream>>>(aggB, fB1, Wg2, aggB, NB);

    // --- final projection into d_out ---
    float* out = (float*)d_out;
    rshn_gemm<false, false><<<gemmA, 256, 0, stream>>>(nullptr, aggA, W_out, out, NA);
    rshn_gemm<false, false><<<gemmB, 256, 0, stream>>>(nullptr, aggB, W_out,
                                                       out + (size_t)NA * D, NB);
}